// GhostLinear_50663434223822
// MI455X (gfx1250) — compile-verified
//
#include <hip/hip_runtime.h>

// GhostLinear on MI455X (gfx1250):
//   Pass 1: W[n][k] = lut[idx[n][k]] * scale[n]  into d_ws (coalesced, 1 gather/weight)
//   Pass 2: out = X @ W^T as f32 WMMA 16x16x4 GEMM, 128x256 block tiles,
//           GLOBAL_LOAD_ASYNC_TO_LDS_B128 double-buffered tile pipeline (ASYNCcnt),
//           both LDS tiles in [row][k] layout, pitch 36 -> conflict-free ds_load_b64 frags.

typedef __attribute__((ext_vector_type(2))) float v2f;
typedef __attribute__((ext_vector_type(4))) float f4;
typedef __attribute__((ext_vector_type(8))) float v8f;
typedef __attribute__((ext_vector_type(4))) int   i4;

#define M_TOTAL 8192   // B*S
#define N_TOTAL 4096   // OUT_F
#define K_TOTAL 4096   // IN_F

#define BM 128
#define BN 256
#define BK 32
#define RP 36                                // LDS row pitch (floats): rows 144B (16B aligned), 36L%64 distinct
#define XTILE (BM * RP)                      // 4608 floats
#define WTILE (BN * RP)                      // 9216 floats
#define BUFSZ (XTILE + WTILE)
#define LDS_BYTES (2 * BUFSZ * 4)            // 110592 B (double buffered; WGP has 320KB)

#define XOFF(bf, e) ((bf) * BUFSZ + (e))
#define WOFF(bf, e) ((bf) * BUFSZ + XTILE + (e))

// ---------------------------------------------------------------------------
// Pass 1: dequantize (fully coalesced int4 read -> 4 gathers -> float4 write)
// ---------------------------------------------------------------------------
__global__ __launch_bounds__(256) void ghost_dequant(
    const int* __restrict__ gidx, const float* __restrict__ lut,
    const float* __restrict__ scale, float* __restrict__ wdq) {
  const int gid = blockIdx.x * 256 + threadIdx.x;
  const int n = gid >> 10;              // 1024 quads per 4096-wide row
  const int k = (gid & 1023) << 2;
  const int base = n * K_TOTAL + k;
  i4 gi = *(const i4*)&gidx[base];
  const float s = scale[n];
  f4 w;
  w.x = lut[gi.x] * s;
  w.y = lut[gi.y] * s;
  w.z = lut[gi.z] * s;
  w.w = lut[gi.w] * s;
  *(f4*)&wdq[base] = w;
}

// ---------------------------------------------------------------------------
// Pass 2: WMMA GEMM. FUSED=true dequantizes W tiles in-kernel (ws too small).
// ---------------------------------------------------------------------------
template <bool FUSED>
__global__ __launch_bounds__(256) void ghost_gemm(
    const float* __restrict__ x,     // [M, K]
    const float* __restrict__ wsrc,  // dequantized W [N, K] (unused if FUSED)
    const int*   __restrict__ gidx,  // [N, K]
    const float* __restrict__ lut,   // [65536]
    const float* __restrict__ scale, // [N]
    float*       __restrict__ out) { // [M, N]
  extern __shared__ float lds[];
  const unsigned ldsBase = __builtin_amdgcn_groupstaticsize();  // dynamic-LDS byte base

  const int t    = threadIdx.x;
  const int lane = t & 31;
  const int wave = t >> 5;            // 8 waves
  const int half = lane >> 4;
  const int l16  = lane & 15;
  const int waveM = wave & 1;         // 2 waves over M -> 64 rows each
  const int waveN = wave >> 1;        // 4 waves over N -> 64 cols each
  const int m0 = blockIdx.x * BM;
  const int n0 = blockIdx.y * BN;

  // cooperative fill mapping: 256 threads; one 16B element per (row-group, quad)
  const int kq  = (t & 7) << 2;       // 0..28
  const int row = t >> 3;             // 0..31

  float sc[8];
  if (FUSED) {
#pragma unroll
    for (int ni = 0; ni < 8; ++ni) sc[ni] = scale[n0 + row + 32 * ni];
  }

  // ---- async tile fills: memory -> LDS with no VGPR staging (ASYNCcnt) ----
  auto issueX = [&](int kb, int bf) {
#pragma unroll
    for (int mi = 0; mi < 4; ++mi) {
      unsigned off = ldsBase + 4u * (unsigned)XOFF(bf, (row + 32 * mi) * RP + kq);
      const float* g = &x[(m0 + row + 32 * mi) * K_TOTAL + kb + kq];
      asm volatile("global_load_async_to_lds_b128 %0, %1, off"
                   :: "v"(off), "v"(g) : "memory");
    }
  };
  auto issueW = [&](int kb, int bf) {  // precomputed-weights path
#pragma unroll
    for (int ni = 0; ni < 8; ++ni) {
      unsigned off = ldsBase + 4u * (unsigned)WOFF(bf, (row + 32 * ni) * RP + kq);
      const float* g = &wsrc[(n0 + row + 32 * ni) * K_TOTAL + kb + kq];
      asm volatile("global_load_async_to_lds_b128 %0, %1, off"
                   :: "v"(off), "v"(g) : "memory");
    }
  };
  auto fillWFused = [&](int kb, int bf) {  // gather path: LUT dequant -> ds_store_b128
#pragma unroll
    for (int ni = 0; ni < 8; ++ni) {
      i4 gi = *(const i4*)&gidx[(n0 + row + 32 * ni) * K_TOTAL + kb + kq];
      f4 w;
      w.x = lut[gi.x] * sc[ni];
      w.y = lut[gi.y] * sc[ni];
      w.z = lut[gi.z] * sc[ni];
      w.w = lut[gi.w] * sc[ni];
      *(f4*)&lds[WOFF(bf, (row + 32 * ni) * RP + kq)] = w;
    }
  };

  v8f acc[4][4] = {};

  auto computeChunk = [&](int bf) {
#pragma unroll
    for (int k0 = 0; k0 < BK; k0 += 4) {
      // A frag (16x4 f32): lanes<16 -> {k0,k0+1}, lanes>=16 -> {k0+2,k0+3}
      v2f a[4], b[4];
#pragma unroll
      for (int mf = 0; mf < 4; ++mf)
        a[mf] = *(const v2f*)&lds[XOFF(bf, (waveM * 64 + mf * 16 + l16) * RP + k0 + 2 * half)];
      // B frag (4x16 f32): same per-lane pairing, read from W's [n][k] rows
#pragma unroll
      for (int nf = 0; nf < 4; ++nf)
        b[nf] = *(const v2f*)&lds[WOFF(bf, (waveN * 64 + nf * 16 + l16) * RP + k0 + 2 * half)];
#pragma unroll
      for (int mf = 0; mf < 4; ++mf)
#pragma unroll
        for (int nf = 0; nf < 4; ++nf)
          acc[mf][nf] = __builtin_amdgcn_wmma_f32_16x16x4_f32(
              false, a[mf], false, b[nf], (short)0, acc[mf][nf], false, false);
    }
  };

  // ---- prologue: fill buffer 0 ----
  issueX(0, 0);
  if (!FUSED) issueW(0, 0); else fillWFused(0, 0);
  asm volatile("s_wait_asynccnt 0" ::: "memory");
  __syncthreads();

  // ---- main pipeline: prefetch next chunk into the other buffer during WMMAs ----
  int bf = 0;
  for (int kb = 0; kb < K_TOTAL; kb += BK) {
    const int kn = kb + BK;
    const bool hasNext = kn < K_TOTAL;
    if (hasNext) {
      issueX(kn, bf ^ 1);
      if (!FUSED) issueW(kn, bf ^ 1); else fillWFused(kn, bf ^ 1);
    }
    computeChunk(bf);
    if (hasNext) {
      asm volatile("s_wait_asynccnt 0" ::: "memory");
      __syncthreads();   // also waits DScnt for the fused ds_store path
      bf ^= 1;
    }
  }

  // ---- epilogue: C/D layout -> VGPR v holds M=v (lanes 0-15) / M=v+8 (16-31) ----
#pragma unroll
  for (int mf = 0; mf < 4; ++mf) {
#pragma unroll
    for (int nf = 0; nf < 4; ++nf) {
      const int colg = n0 + waveN * 64 + nf * 16 + l16;
#pragma unroll
      for (int v = 0; v < 8; ++v) {
        const int rowg = m0 + waveM * 64 + mf * 16 + v + 8 * half;
        out[rowg * N_TOTAL + colg] = acc[mf][nf][v];
      }
    }
  }
}

// ---------------------------------------------------------------------------
extern "C" void kernel_launch(void* const* d_in, const int* in_sizes, int n_in,
                              void* d_out, int out_size, void* d_ws, size_t ws_size,
                              hipStream_t stream) {
  const float* x     = (const float*)d_in[0];
  const int*   gidx  = (const int*)d_in[1];
  const float* lut   = (const float*)d_in[2];
  const float* scale = (const float*)d_in[3];
  float*       out   = (float*)d_out;

  const size_t need = (size_t)N_TOTAL * K_TOTAL * sizeof(float);  // 64 MB
  dim3 grid(M_TOTAL / BM, N_TOTAL / BN);                          // 64 x 16 = 1024 WGs

  if (ws_size >= need) {
    float* wdq = (float*)d_ws;
    (void)hipFuncSetAttribute((const void*)ghost_gemm<false>,
                              hipFuncAttributeMaxDynamicSharedMemorySize, LDS_BYTES);
    ghost_dequant<<<(N_TOTAL * (K_TOTAL / 4)) / 256, 256, 0, stream>>>(gidx, lut, scale, wdq);
    ghost_gemm<false><<<grid, 256, LDS_BYTES, stream>>>(x, wdq, gidx, lut, scale, out);
  } else {
    (void)hipFuncSetAttribute((const void*)ghost_gemm<true>,
                              hipFuncAttributeMaxDynamicSharedMemorySize, LDS_BYTES);
    ghost_gemm<true><<<grid, 256, LDS_BYTES, stream>>>(x, nullptr, gidx, lut, scale, out);
  }
}